// DigitCapsLayer_62182536511683
// MI455X (gfx1250) — compile-verified
//
#include <hip/hip_runtime.h>
#include <math.h>

// Problem constants
#define BATCH 256
#define IN_CH 1152
#define IN_U  8
#define OUT_CH 10
#define OUT_U 16
#define EPOCHS 3

typedef __attribute__((ext_vector_type(2))) float v2f;
typedef __attribute__((ext_vector_type(8))) float v8f;

__device__ __forceinline__ v8f wmma_f32_4(v2f a, v2f b, v8f c) {
    // D = A(16x4 f32) x B(4x16 f32) + C(16x16 f32)
    return __builtin_amdgcn_wmma_f32_16x16x4_f32(
        /*neg_a=*/false, a, /*neg_b=*/false, b,
        /*c_mod=*/(short)0, c, /*reuse_a=*/false, /*reuse_b=*/false);
}

// ---------------------------------------------------------------------------
// zero a float buffer
__global__ void zero_kernel(float* __restrict__ p, int n) {
    int t = blockIdx.x * blockDim.x + threadIdx.x;
    if (t < n) p[t] = 0.0f;
}

// ---------------------------------------------------------------------------
// c[i,:] = softmax(b[i,:]) over OUT_CH
__global__ void softmax_kernel(const float* __restrict__ b, float* __restrict__ c) {
    int i = blockIdx.x * blockDim.x + threadIdx.x;
    if (i >= IN_CH) return;
    float mx = -3.402823466e38f;
    float e[OUT_CH];
    #pragma unroll
    for (int o = 0; o < OUT_CH; ++o) mx = fmaxf(mx, b[i * OUT_CH + o]);
    float sum = 0.0f;
    #pragma unroll
    for (int o = 0; o < OUT_CH; ++o) { e[o] = expf(b[i * OUT_CH + o] - mx); sum += e[o]; }
    float inv = 1.0f / sum;
    #pragma unroll
    for (int o = 0; o < OUT_CH; ++o) c[i * OUT_CH + o] = e[o] * inv;
}

// ---------------------------------------------------------------------------
// s[b,o,u] = sum_{i,k} c[i,o] * W[i,o,u,k] * inp[b,i,k]
// One wave per (batch-tile of 16, o, i-chunk of 72). M=batch, N=u, K=(i,k).
#define ICHUNK 72
#define NCHUNK (IN_CH / ICHUNK)   // 16
__global__ __launch_bounds__(32)
void s_gemm_kernel(const float* __restrict__ inp, const float* __restrict__ W,
                   const float* __restrict__ c, float* __restrict__ s) {
    const int l  = threadIdx.x;
    const int m  = l & 15;     // A: batch row in tile ; B: u ; D: u column
    const int hi = l >> 4;     // K-pair selector
    const int bt = blockIdx.x; // 0..15  batch tile
    const int o  = blockIdx.y; // 0..9
    const int i0 = blockIdx.z * ICHUNK;

    const int brow = bt * 16 + m;
    const float* arow = inp + (size_t)brow * IN_CH * IN_U;

    v8f acc = {};
    for (int ii = 0; ii < ICHUNK; ++ii) {
        const int i = i0 + ii;
        const float cio = c[i * OUT_CH + o];
        const float* ap = arow + (size_t)i * IN_U + 2 * hi;
        const float* wp = W + (((size_t)i * OUT_CH + o) * OUT_U + m) * IN_U + 2 * hi;
        const float2 a0 = *(const float2*)(ap);
        const float2 a1 = *(const float2*)(ap + 4);
        const float2 w0 = *(const float2*)(wp);
        const float2 w1 = *(const float2*)(wp + 4);
        v2f A0; A0[0] = a0.x;       A0[1] = a0.y;
        v2f B0; B0[0] = cio * w0.x; B0[1] = cio * w0.y;
        acc = wmma_f32_4(A0, B0, acc);
        v2f A1; A1[0] = a1.x;       A1[1] = a1.y;
        v2f B1; B1[0] = cio * w1.x; B1[1] = cio * w1.y;
        acc = wmma_f32_4(A1, B1, acc);
    }
    // D: lane holds u=m, rows j+8*hi of the batch tile
    #pragma unroll
    for (int j = 0; j < 8; ++j) {
        const int br = bt * 16 + j + 8 * hi;
        unsafeAtomicAdd(&s[((size_t)br * OUT_CH + o) * OUT_U + m], acc[j]);
    }
}

// ---------------------------------------------------------------------------
// v[b,o,:] = squash(s[b,o,:]) ; v is written into d_out every epoch
__global__ void squash_kernel(const float* __restrict__ s, float* __restrict__ v) {
    int t = blockIdx.x * blockDim.x + threadIdx.x; // (b*10+o)
    if (t >= BATCH * OUT_CH) return;
    const float* sp = s + (size_t)t * OUT_U;
    float sqn = 0.0f;
    #pragma unroll
    for (int u = 0; u < OUT_U; ++u) sqn += sp[u] * sp[u];
    const float scale = sqn / ((1.0f + sqn) * sqrtf(sqn + 1e-9f));
    #pragma unroll
    for (int u = 0; u < OUT_U; ++u) v[(size_t)t * OUT_U + u] = sp[u] * scale;
}

// ---------------------------------------------------------------------------
// b[i,o] += (1/B) * sum_{k,u} W[i,o,u,k] * P[i,k,u],
//   P[i,k,u] = sum_b inp[b,i,k] * v[b,o,u]
// One wave per (i-pair, o). M=(i,k) stacked (2 i's x 8 k), N=u, K=batch.
__global__ __launch_bounds__(32)
void delta_kernel(const float* __restrict__ inp, const float* __restrict__ W,
                  const float* __restrict__ v, float* __restrict__ b) {
    const int l  = threadIdx.x;
    const int m  = l & 15;
    const int hi = l >> 4;
    const int i0 = blockIdx.x * 2;
    const int o  = blockIdx.y;

    const int i_a = i0 + (m >> 3);  // A-row i
    const int k_a = m & 7;          // A-row k
    const float* ap = inp + (size_t)i_a * IN_U + k_a;          // + b*9216
    const float* vp = v + (size_t)o * OUT_U + m;               // + b*160

    v8f acc = {};
    for (int b0 = 0; b0 < BATCH; b0 += 4) {
        const int t0 = b0 + 2 * hi;
        v2f A;  A[0]  = ap[(size_t)t0 * (IN_CH * IN_U)];
                A[1]  = ap[(size_t)(t0 + 1) * (IN_CH * IN_U)];
        v2f Bv; Bv[0] = vp[(size_t)t0 * (OUT_CH * OUT_U)];
                Bv[1] = vp[(size_t)(t0 + 1) * (OUT_CH * OUT_U)];
        acc = wmma_f32_4(A, Bv, acc);
    }
    // D: lane holds u=m; rows j+8*hi => i_d = i0+hi, k = j
    const int i_d = i0 + hi;
    const float* wp = W + (((size_t)i_d * OUT_CH + o) * OUT_U + m) * IN_U;
    float r = 0.0f;
    #pragma unroll
    for (int j = 0; j < 8; ++j) r += acc[j] * wp[j];
    // reduce over u (16 lanes per half-wave; masks 1..8 never cross halves)
    #pragma unroll
    for (int off = 1; off < 16; off <<= 1) r += __shfl_xor(r, off, 32);
    if (m == 0) b[(size_t)i_d * OUT_CH + o] += r * (1.0f / (float)BATCH);
}

// ---------------------------------------------------------------------------
extern "C" void kernel_launch(void* const* d_in, const int* in_sizes, int n_in,
                              void* d_out, int out_size, void* d_ws, size_t ws_size,
                              hipStream_t stream) {
    const float* inp = (const float*)d_in[0]; // [256,1152,8]
    const float* W   = (const float*)d_in[1]; // [1152,10,16,8]
    float* vout = (float*)d_out;              // [256,10,16]

    float* ws   = (float*)d_ws;
    float* bbuf = ws;                         // [1152,10]
    float* cbuf = ws + IN_CH * OUT_CH;        // [1152,10]
    float* sbuf = ws + 2 * IN_CH * OUT_CH;    // [256,10,16]

    const int NB = IN_CH * OUT_CH;            // 11520
    const int NS = BATCH * OUT_CH * OUT_U;    // 40960

    zero_kernel<<<(NB + 255) / 256, 256, 0, stream>>>(bbuf, NB);

    for (int e = 0; e < EPOCHS; ++e) {
        softmax_kernel<<<(IN_CH + 255) / 256, 256, 0, stream>>>(bbuf, cbuf);
        zero_kernel<<<(NS + 255) / 256, 256, 0, stream>>>(sbuf, NS);
        s_gemm_kernel<<<dim3(BATCH / 16, OUT_CH, NCHUNK), 32, 0, stream>>>(inp, W, cbuf, sbuf);
        squash_kernel<<<(BATCH * OUT_CH + 255) / 256, 256, 0, stream>>>(sbuf, vout);
        if (e < EPOCHS - 1) {
            delta_kernel<<<dim3(IN_CH / 2, OUT_CH), 32, 0, stream>>>(inp, W, vout, bbuf);
        }
    }
}